// ContinuityLoss_73100343378533
// MI455X (gfx1250) — compile-verified
//
#include <hip/hip_runtime.h>

typedef float v2f __attribute__((ext_vector_type(2)));
typedef float v4f __attribute__((ext_vector_type(4)));
typedef float v8f __attribute__((ext_vector_type(8)));

#define TILE 16
#define LAMBDA_CONT 0.01f
// -1/(2*ln2): exp(-d2/2) == exp2(d2 * NEG_HALF_LOG2E)
#define NEG_HALF_LOG2E (-0.72134752044448170367996234050095f)

// ---------------- Kernel 0: row squared norms ----------------
__global__ void sqnorm_kernel(const float* __restrict__ pts,
                              const float* __restrict__ outs,
                              float* __restrict__ sqp,
                              float* __restrict__ sqo, int n) {
    int i = blockIdx.x * blockDim.x + threadIdx.x;
    if (i >= n) return;
    float px = pts[2 * i], py = pts[2 * i + 1];
    sqp[i] = px * px + py * py;
    v4f a = *(const v4f*)(outs + 8 * i);
    v4f b = *(const v4f*)(outs + 8 * i + 4);
    sqo[i] = a.x * a.x + a.y * a.y + a.z * a.z + a.w * a.w +
             b.x * b.x + b.y * b.y + b.z * b.z + b.w * b.w;
}

// ---------------- Kernel 1: tiled pairwise via WMMA ----------------
// Block = 1 i-tile (16 rows), 256 threads = 8 waves. Wave w sweeps j-tiles
// jt = w, w+8, ... Each 16x16 tile: Gram(P) via 1 WMMA (K=2 padded to 4),
// Gram(O) via 2 chained WMMAs (K=8). Then exp/sqrt on the f32 accumulators.
__global__ __launch_bounds__(256) void pairtile_kernel(
    const float* __restrict__ pts, const float* __restrict__ outs,
    const float* __restrict__ sqp, const float* __restrict__ sqo,
    float* __restrict__ partials, int n) {
    const int lane = threadIdx.x & 31;
    const int wave = threadIdx.x >> 5;
    const int half = lane >> 4;    // 0: K=0,1 (A) / M=0..7 (C); 1: K=2,3 / M=8..15
    const int l16  = lane & 15;
    const int nt   = n / TILE;
    const int i0   = blockIdx.x * TILE;
    const int irow = i0 + l16;

    // ---- A operands (held for whole kernel) ----
    // points A (16x4, K=2 real + 2 zero-pad): lanes 16-31 carry the pad Ks.
    v2f ap; ap.x = 0.0f; ap.y = 0.0f;
    if (half == 0) { ap.x = pts[2 * irow]; ap.y = pts[2 * irow + 1]; }
    // outputs A split into two K=4 chunks (cols 0..3 and 4..7)
    v4f olo = *(const v4f*)(outs + 8 * irow);
    v4f ohi = *(const v4f*)(outs + 8 * irow + 4);
    v2f ao1, ao2;
    if (half == 0) { ao1.x = olo.x; ao1.y = olo.y; ao2.x = ohi.x; ao2.y = ohi.y; }
    else           { ao1.x = olo.z; ao1.y = olo.w; ao2.x = ohi.z; ao2.y = ohi.w; }

    // per-lane row squared norms for C rows: row(r) = i0 + 8*half + r, r=0..7
    float sqpi[8], sqoi[8];
    {
        const float* p = sqp + i0 + 8 * half;
        const float* q = sqo + i0 + 8 * half;
#pragma unroll
        for (int r = 0; r < 8; ++r) { sqpi[r] = p[r]; sqoi[r] = q[r]; }
    }

    float acc = 0.0f;
    for (int jt = wave; jt < nt; jt += 8) {
        const int j = jt * TILE + l16;   // this lane's column row-index
        // ---- B operands for this j-tile ----
        float bpx = pts[2 * j], bpy = pts[2 * j + 1];
        v2f bp; bp.x = 0.0f; bp.y = 0.0f;
        if (half == 0) { bp.x = bpx; bp.y = bpy; }
        v4f blo = *(const v4f*)(outs + 8 * j);
        v4f bhi = *(const v4f*)(outs + 8 * j + 4);
        v2f bo1, bo2;
        if (half == 0) { bo1.x = blo.x; bo1.y = blo.y; bo2.x = bhi.x; bo2.y = bhi.y; }
        else           { bo1.x = blo.z; bo1.y = blo.w; bo2.x = bhi.z; bo2.y = bhi.w; }
        const float sqp_j = sqp[j];
        const float sqo_j = sqo[j];

        // ---- Gram tiles via WMMA (D = A*B + C), f32 16x16x4 ----
        v8f cp = {};
        cp = __builtin_amdgcn_wmma_f32_16x16x4_f32(false, ap, false, bp,
                                                   (short)0, cp, false, false);
        v8f co = {};
        co = __builtin_amdgcn_wmma_f32_16x16x4_f32(false, ao1, false, bo1,
                                                   (short)0, co, false, false);
        co = __builtin_amdgcn_wmma_f32_16x16x4_f32(false, ao2, false, bo2,
                                                   (short)0, co, false, false);

        // ---- elementwise: w = exp(-d2p/2), term = w * sqrt(d2o) ----
#pragma unroll
        for (int r = 0; r < 8; ++r) {
            const int ir = i0 + 8 * half + r;
            float d2p = fmaxf(sqpi[r] + sqp_j - 2.0f * cp[r], 0.0f);
            float w = __builtin_amdgcn_exp2f(d2p * NEG_HALF_LOG2E); // v_exp_f32
            if (ir == j) w = 0.0f;                                  // zero diagonal
            float d2o = fmaxf(sqoi[r] + sqo_j - 2.0f * co[r], 0.0f);
            acc = fmaf(w, __builtin_amdgcn_sqrtf(d2o), acc);        // v_sqrt_f32
        }
    }

    // ---- deterministic block reduction ----
    __shared__ float red[256];
    red[threadIdx.x] = acc;
    __syncthreads();
#pragma unroll
    for (int s = 128; s > 0; s >>= 1) {
        if (threadIdx.x < s) red[threadIdx.x] += red[threadIdx.x + s];
        __syncthreads();
    }
    if (threadIdx.x == 0) partials[blockIdx.x] = red[0];
}

// ---------------- Kernel 2: finalize ----------------
__global__ void finalize_kernel(const float* __restrict__ partials, int cnt,
                                float n_f, float* __restrict__ out) {
    __shared__ float red[256];
    float a = 0.0f;
    for (int i = threadIdx.x; i < cnt; i += 256) a += partials[i];
    red[threadIdx.x] = a;
    __syncthreads();
#pragma unroll
    for (int s = 128; s > 0; s >>= 1) {
        if (threadIdx.x < s) red[threadIdx.x] += red[threadIdx.x + s];
        __syncthreads();
    }
    if (threadIdx.x == 0)
        out[0] = red[0] * (LAMBDA_CONT / (n_f * (n_f - 1.0f)));
}

extern "C" void kernel_launch(void* const* d_in, const int* in_sizes, int n_in,
                              void* d_out, int out_size, void* d_ws, size_t ws_size,
                              hipStream_t stream) {
    const float* pts  = (const float*)d_in[0];   // [n,2] f32
    const float* outs = (const float*)d_in[1];   // [n,8] f32
    float* out = (float*)d_out;
    const int n  = in_sizes[0] / 2;              // 8192
    const int nt = n / TILE;                     // 512

    // workspace layout (floats): sqp[n] | sqo[n] | partials[nt]
    float* sqp      = (float*)d_ws;
    float* sqo      = sqp + n;
    float* partials = sqo + n;

    sqnorm_kernel<<<(n + 255) / 256, 256, 0, stream>>>(pts, outs, sqp, sqo, n);
    pairtile_kernel<<<nt, 256, 0, stream>>>(pts, outs, sqp, sqo, partials, n);
    finalize_kernel<<<1, 256, 0, stream>>>(partials, nt, (float)n, out);
}